// HMLSTM_41540923687174
// MI455X (gfx1250) — compile-verified
//
#include <hip/hip_runtime.h>
#include <math.h>

typedef float v2f __attribute__((ext_vector_type(2)));
typedef float v4f __attribute__((ext_vector_type(4)));
typedef float v8f __attribute__((ext_vector_type(8)));

#define T_STEPS 256
#define BB      64
#define HH      384
#define GG      1537            // 4H+1 rows
#define HB      (HH*BB)         // 24576
#define KMAX    1152
#define PBSZ    (KMAX*BB)       // 73728 floats per packed-B buffer

#define NBLOCKS  25             // 24 gate-blocks + 1 boundary block
#define NTHREADS 256
#define NTOT     (NBLOCKS*NTHREADS)

// workspace layout (floats): pB[3][2] then state
#define OFF_PB    0                       // 6 buffers of PBSZ = 442368
#define OFF_H     (6*PBSZ)                // 442368
#define OFF_C     (OFF_H + 3*HB)          // 516096
#define OFF_Z     (OFF_C + 3*HB)          // 589824  zbuf[2][3][64]
#define OFF_BAR   (OFF_Z + 2*3*BB)        // 590208  2 ints
#define WS_FLOATS (OFF_BAR + 2)

// element (k,n) of B -> packed order: one b128/lane feeds both n-tiles of a half
// chunk kc=k>>2, nhalf=n>>5, lane=((k>>1)&1)*16+(n&15), slot=((n>>4)&1)*2+(k&1)
__device__ __forceinline__ int frag_idx(int k, int n) {
    int kc = k >> 2, nh = n >> 5;
    int lane = (((k >> 1) & 1) << 4) | (n & 15);
    return ((kc * 2 + nh) * 32 + lane) * 4 + (((n >> 4) & 1) << 1) + (k & 1);
}

__device__ __forceinline__ float sigm(float x) { return 1.0f / (1.0f + expf(-x)); }

__device__ __forceinline__ void gbar(int* bar) {
    __threadfence();
    __syncthreads();
    if (threadIdx.x == 0) {
        int g = __hip_atomic_load(&bar[1], __ATOMIC_RELAXED, __HIP_MEMORY_SCOPE_AGENT);
        int a = __hip_atomic_fetch_add(&bar[0], 1, __ATOMIC_ACQ_REL, __HIP_MEMORY_SCOPE_AGENT);
        if (a == NBLOCKS - 1) {
            __hip_atomic_store(&bar[0], 0, __ATOMIC_RELAXED, __HIP_MEMORY_SCOPE_AGENT);
            __hip_atomic_store(&bar[1], g + 1, __ATOMIC_RELEASE, __HIP_MEMORY_SCOPE_AGENT);
        } else {
            while (__hip_atomic_load(&bar[1], __ATOMIC_ACQUIRE, __HIP_MEMORY_SCOPE_AGENT) == g) {
                __builtin_amdgcn_s_sleep(1);
            }
        }
    }
    __syncthreads();
    __threadfence();
}

__global__ void __launch_bounds__(NTHREADS)
hmlstm_init(float* ws) {
    int stride = gridDim.x * blockDim.x;
    for (int i = blockIdx.x * blockDim.x + threadIdx.x; i < WS_FLOATS; i += stride)
        ws[i] = 0.0f;
}

__global__ void __launch_bounds__(NTHREADS)
hmlstm_persistent(const float* __restrict__ x,
                  const float* __restrict__ W0, const float* __restrict__ U110,
                  const float* __restrict__ U210, const float* __restrict__ b0,
                  const float* __restrict__ W1, const float* __restrict__ U111,
                  const float* __restrict__ U211, const float* __restrict__ b1,
                  const float* __restrict__ W2, const float* __restrict__ U112,
                  const float* __restrict__ b2,
                  float* __restrict__ out, float* __restrict__ ws) {
    __shared__ float Slds0[4 * 16 * 64];    // S0 = W @ h_bottom      (16KB)
    __shared__ float Slds1[4 * 16 * 64];    // S1 = U11@h + U21@h_top (16KB)

    float* pbuf = ws + OFF_PB;
    float* h    = ws + OFF_H;
    float* c    = ws + OFF_C;
    float* zbuf = ws + OFF_Z;
    int*   bar  = (int*)(ws + OFF_BAR);

    const float* Wt0[3] = {W0, U110, U210};
    const float* Wt1[3] = {W1, U111, U211};
    const float* Wt2[3] = {W2, U112, U112};
    const float* const* Wt[3] = {Wt0, Wt1, Wt2};
    const float* bias[3] = {b0, b1, b2};

    const int tid  = blockIdx.x * NTHREADS + threadIdx.x;
    const int lane = threadIdx.x & 31;
    const int lw   = threadIdx.x >> 5;          // wave in block (wave32)
    const bool gateblk = (blockIdx.x < 24);

    // gemm job: strip of 16 rows (gate gl), nhalf of 32 cols
    int strip, nh, gl;
    bool active;
    if (gateblk) { strip = blockIdx.x + 24 * (lw >> 1); nh = lw & 1; gl = lw >> 1; active = true; }
    else         { strip = 96; nh = lw & 1; gl = 0; active = (lw < 2); }
    const int m    = strip * 16 + (lane & 15);
    const int mrow = (m < GG) ? m : (GG - 1);   // clamp; garbage rows land in unread pad
    const int koff = (lane >> 4) << 1;          // 0 or 2

    // preload per-thread bias values (j-invariant over time)
    float bpre[3][4][4];
    float bbnd[3];
    if (gateblk) {
#pragma unroll
        for (int l = 0; l < 3; ++l)
#pragma unroll
            for (int k = 0; k < 4; ++k) {
                int e = threadIdx.x + k * NTHREADS;
                int j = blockIdx.x * 16 + (e >> 6);
                bpre[l][k][0] = bias[l][j];
                bpre[l][k][1] = bias[l][HH + j];
                bpre[l][k][2] = bias[l][2 * HH + j];
                bpre[l][k][3] = bias[l][3 * HH + j];
            }
    } else {
#pragma unroll
        for (int l = 0; l < 3; ++l) bbnd[l] = bias[l][4 * HH];
    }

    // prologue: region0 of pB[layer0][par0] = x[0]; everything else zeroed by init
    {
        float* pb00 = pbuf;                      // (l=0,par=0)
        for (int e = tid; e < HB; e += NTOT) {
            int j = e >> 6, b = e & 63;
            pb00[frag_idx(j, b)] = x[(size_t)j * BB + b];
        }
    }
    gbar(bar);

    for (int t = 0; t < T_STEPS; ++t) {
        const int rd = t & 1, wr = (t + 1) & 1;
#pragma unroll
        for (int l = 0; l < 3; ++l) {
            const float* pRd = pbuf + (size_t)(l * 2 + rd) * PBSZ;
            // ===== GEMM: S0 += W@hb (region0), S1 += U11@h + U21@htop =======
            if (active) {
                v8f p0 = {0.f,0.f,0.f,0.f,0.f,0.f,0.f,0.f};
                v8f p1 = {0.f,0.f,0.f,0.f,0.f,0.f,0.f,0.f};
                v8f q0 = {0.f,0.f,0.f,0.f,0.f,0.f,0.f,0.f};
                v8f q1 = {0.f,0.f,0.f,0.f,0.f,0.f,0.f,0.f};
                {   // region 0 -> S0
                    const float* Ab = Wt[l][0] + (size_t)mrow * 384 + koff;
                    const float* Bb = pRd + (size_t)nh * 128 + (size_t)lane * 4;
#pragma unroll 8
                    for (int kc = 0; kc < 96; ++kc) {
                        v2f a  = *(const v2f*)(Ab + kc * 4);
                        v4f bv = *(const v4f*)(Bb + kc * 256);
                        v2f bf0 = {bv.x, bv.y};
                        v2f bf1 = {bv.z, bv.w};
                        p0 = __builtin_amdgcn_wmma_f32_16x16x4_f32(
                                 false, a, false, bf0, (short)0, p0, false, false);
                        p1 = __builtin_amdgcn_wmma_f32_16x16x4_f32(
                                 false, a, false, bf1, (short)0, p1, false, false);
                    }
                }
                {   // region 1 -> S1
                    const float* Ab = Wt[l][1] + (size_t)mrow * 384 + koff;
                    const float* Bb = pRd + 24576 + (size_t)nh * 128 + (size_t)lane * 4;
#pragma unroll 8
                    for (int kc = 0; kc < 96; ++kc) {
                        v2f a  = *(const v2f*)(Ab + kc * 4);
                        v4f bv = *(const v4f*)(Bb + kc * 256);
                        v2f bf0 = {bv.x, bv.y};
                        v2f bf1 = {bv.z, bv.w};
                        q0 = __builtin_amdgcn_wmma_f32_16x16x4_f32(
                                 false, a, false, bf0, (short)0, q0, false, false);
                        q1 = __builtin_amdgcn_wmma_f32_16x16x4_f32(
                                 false, a, false, bf1, (short)0, q1, false, false);
                    }
                }
                if (l < 2) {   // region 2 -> S1
                    const float* Ab = Wt[l][2] + (size_t)mrow * 384 + koff;
                    const float* Bb = pRd + 49152 + (size_t)nh * 128 + (size_t)lane * 4;
#pragma unroll 8
                    for (int kc = 0; kc < 96; ++kc) {
                        v2f a  = *(const v2f*)(Ab + kc * 4);
                        v4f bv = *(const v4f*)(Bb + kc * 256);
                        v2f bf0 = {bv.x, bv.y};
                        v2f bf1 = {bv.z, bv.w};
                        q0 = __builtin_amdgcn_wmma_f32_16x16x4_f32(
                                 false, a, false, bf0, (short)0, q0, false, false);
                        q1 = __builtin_amdgcn_wmma_f32_16x16x4_f32(
                                 false, a, false, bf1, (short)0, q1, false, false);
                    }
                }
                const int nb = nh * 32 + (lane & 15);
                const int rl = (lane >> 4) << 3;
#pragma unroll
                for (int i = 0; i < 8; ++i) {
                    int base = (gl * 16 + rl + i) * 64 + nb;
                    Slds0[base]      = p0[i];
                    Slds0[base + 16] = p1[i];
                    Slds1[base]      = q0[i];
                    Slds1[base + 16] = q1[i];
                }
            }
            __syncthreads();
            // ===== pointwise: S = S0 + z*S1 + bias; fused producer-side pack ==
            if (gateblk) {
                float* hl = h + l * HB;
                float* cl = c + l * HB;
#pragma unroll
                for (int k2 = 0; k2 < 4; ++k2) {
                    int e  = threadIdx.x + k2 * NTHREADS;
                    int jl = e >> 6, b = e & 63;
                    int j  = blockIdx.x * 16 + jl;
                    float z  = zbuf[rd * 192 + l * 64 + b];
                    float sf = Slds0[(0*16+jl)*64+b] + z * Slds1[(0*16+jl)*64+b] + bpre[l][k2][0];
                    float si = Slds0[(1*16+jl)*64+b] + z * Slds1[(1*16+jl)*64+b] + bpre[l][k2][1];
                    float so = Slds0[(2*16+jl)*64+b] + z * Slds1[(2*16+jl)*64+b] + bpre[l][k2][2];
                    float sg = Slds0[(3*16+jl)*64+b] + z * Slds1[(3*16+jl)*64+b] + bpre[l][k2][3];
                    float f  = sigm(sf), ii = sigm(si), o = sigm(so), g = tanhf(sg);
                    float zb = (l == 0) ? 1.0f : zbuf[wr * 192 + (l - 1) * 64 + b];
                    int   ge = j * 64 + b;
                    float cold = cl[ge], hold = hl[ge];
                    float omz = 1.0f - z, ig = ii * g;
                    float cn = z * ig + omz * (1.0f - zb) * cold + omz * zb * (f * cold + ig);
                    float th = tanhf(cn);
                    float hn = z * o * th + omz * (1.0f - zb) * hold + omz * zb * o * th;
                    cl[ge] = cn; hl[ge] = hn;
                    out[((size_t)t * 3 + l) * HB + ge] = hn;
                    // fused packing of raw h into consumer buffers
                    if (l == 0) {
                        pbuf[(size_t)(2 + rd) * PBSZ + frag_idx(j, b)]           = hn; // pB[1] r0, t
                        pbuf[(size_t)(0 + wr) * PBSZ + frag_idx(HH + j, b)]      = hn; // pB[0] r1, t+1
                    } else if (l == 1) {
                        pbuf[(size_t)(4 + rd) * PBSZ + frag_idx(j, b)]           = hn; // pB[2] r0, t
                        pbuf[(size_t)(2 + wr) * PBSZ + frag_idx(HH + j, b)]      = hn; // pB[1] r1, t+1
                        pbuf[(size_t)(0 + wr) * PBSZ + frag_idx(2 * HH + j, b)]  = hn; // pB[0] r2, t+1
                    } else {
                        pbuf[(size_t)(4 + wr) * PBSZ + frag_idx(HH + j, b)]      = hn; // pB[2] r1, t+1
                        pbuf[(size_t)(2 + wr) * PBSZ + frag_idx(2 * HH + j, b)]  = hn; // pB[1] r2, t+1
                        if (t + 1 < T_STEPS)                                          // pB[0] r0, t+1
                            pbuf[(size_t)(0 + wr) * PBSZ + frag_idx(j, b)] =
                                x[((size_t)(t + 1) * HH + j) * BB + b];
                    }
                }
            } else {
                // boundary row 4H (local row 0): z_new = STE fwd = (s > 0)
                for (int e = threadIdx.x; e < BB; e += NTHREADS) {
                    float zp = zbuf[rd * 192 + l * 64 + e];
                    float s4 = Slds0[e] + zp * Slds1[e] + bbnd[l];
                    zbuf[wr * 192 + l * 64 + e] = (s4 > 0.0f) ? 1.0f : 0.0f;
                }
            }
            gbar(bar);
        }
    }
}

extern "C" void kernel_launch(void* const* d_in, const int* in_sizes, int n_in,
                              void* d_out, int out_size, void* d_ws, size_t ws_size,
                              hipStream_t stream) {
    (void)in_sizes; (void)n_in; (void)out_size; (void)ws_size;
    const float* xx   = (const float*)d_in[0];
    const float* W0   = (const float*)d_in[1];
    const float* U110 = (const float*)d_in[2];
    const float* U210 = (const float*)d_in[3];
    const float* b0   = (const float*)d_in[4];
    const float* W1   = (const float*)d_in[5];
    const float* U111 = (const float*)d_in[6];
    const float* U211 = (const float*)d_in[7];
    const float* b1   = (const float*)d_in[8];
    const float* W2   = (const float*)d_in[9];
    const float* U112 = (const float*)d_in[10];
    const float* b2   = (const float*)d_in[11];
    float* out = (float*)d_out;
    float* ws  = (float*)d_ws;

    hmlstm_init<<<128, NTHREADS, 0, stream>>>(ws);
    hmlstm_persistent<<<NBLOCKS, NTHREADS, 0, stream>>>(
        xx, W0, U110, U210, b0, W1, U111, U211, b1, W2, U112, b2, out, ws);
}